// GCN_83734682403304
// MI455X (gfx1250) — compile-verified
//
#include <hip/hip_runtime.h>

typedef __attribute__((ext_vector_type(2))) float v2f;
typedef __attribute__((ext_vector_type(8))) float v8f;

// ---------------- utility kernels ----------------
__global__ void fill_kernel(float* __restrict__ p, float v, int n) {
    int i = blockIdx.x * blockDim.x + threadIdx.x;
    int stride = gridDim.x * blockDim.x;
    for (; i < n; i += stride) p[i] = v;
}

__global__ void degree_kernel(const long long* __restrict__ dst, float* __restrict__ deg, int nE) {
    int e = blockIdx.x * blockDim.x + threadIdx.x;
    if (e < nE) atomicAdd(&deg[(int)dst[e]], 1.0f);
}

__global__ void rsqrt_kernel(float* __restrict__ p, int n) {
    int i = blockIdx.x * blockDim.x + threadIdx.x;
    if (i < n) p[i] = rsqrtf(p[i]);
}

// ---------------- WMMA fp32 GEMM: H[M,NOUT] = A[M,K] @ W[K,NOUT] ----------------
// One 16x16 output tile per wave32; V_WMMA_F32_16X16X4_F32, K unrolled in steps of 4.
template <int K, int NOUT>
__global__ void gemm_wmma_f32(const float* __restrict__ A, const float* __restrict__ W,
                              float* __restrict__ H, int M, int mtiles) {
    constexpr int NTILES = NOUT / 16;
    const int wave = threadIdx.x >> 5;
    const int tile = blockIdx.x * (blockDim.x >> 5) + wave;
    if (tile >= mtiles * NTILES) return;   // wave-uniform exit: EXEC stays all-ones for WMMA

    const int m0 = (tile / NTILES) * 16;
    const int n0 = (tile % NTILES) * 16;
    const int lane = threadIdx.x & 31;
    const int half = lane >> 4;    // selects K pair (A/B) and M block (C/D)
    const int mn   = lane & 15;    // M index for A, N index for B/C/D

    int mrow = m0 + mn;
    if (mrow > M - 1) mrow = M - 1;        // clamp loads on ragged last tile
    const float* __restrict__ arow = A + (long long)mrow * K;

    v8f c = {0.f, 0.f, 0.f, 0.f, 0.f, 0.f, 0.f, 0.f};

#pragma unroll
    for (int k = 0; k < K; k += 4) {
        const int ka = k + 2 * half;
        v2f a, b;
        a.x = arow[ka];
        a.y = arow[ka + 1];
        b.x = W[(long long)ka * NOUT + n0 + mn];
        b.y = W[(long long)(ka + 1) * NOUT + n0 + mn];
        // (neg_a, A, neg_b, B, c_mod, C, reuse_a, reuse_b)
        c = __builtin_amdgcn_wmma_f32_16x16x4_f32(false, a, false, b, (short)0, c, false, false);
    }

#pragma unroll
    for (int v = 0; v < 8; ++v) {
        const int m = m0 + v + 8 * half;   // C/D layout: VGPR v, lanes 16-31 hold M=v+8
        if (m < M) H[(long long)m * NOUT + n0 + mn] = c[v];
    }
}

// ---------------- edge scatter: out[dst] += dinv[src]*dinv[dst] * h[src] ----------------
// One thread per (edge, float4 chunk); 4 fp32 global atomics per thread (L2-resident).
template <int F4, int F>
__global__ void edge_scatter(const long long* __restrict__ src, const long long* __restrict__ dst,
                             const float* __restrict__ dinv, const float* __restrict__ h,
                             float* __restrict__ out, int nE) {
    const int gid = blockIdx.x * blockDim.x + threadIdx.x;
    if (gid >= nE * F4) return;
    const int e = gid / F4;
    const int c = gid % F4;
    const int s = (int)src[e];
    const int d = (int)dst[e];
    const float norm = dinv[s] * dinv[d];
    const float4 hv = ((const float4*)(h + (long long)s * F))[c];
    float* o = out + (long long)d * F + c * 4;
    atomicAdd(o + 0, hv.x * norm);
    atomicAdd(o + 1, hv.y * norm);
    atomicAdd(o + 2, hv.z * norm);
    atomicAdd(o + 3, hv.w * norm);
}

// ---------------- self-loop + bias (+ optional ReLU), in place on agg ----------------
template <int F, bool RELU>
__global__ void self_bias(float* __restrict__ agg, const float* __restrict__ h,
                          const float* __restrict__ dinv, const float* __restrict__ b, int N) {
    constexpr int F4 = F / 4;
    const int gid = blockIdx.x * blockDim.x + threadIdx.x;
    if (gid >= N * F4) return;
    const int i = gid / F4;
    const int c = gid % F4;
    const float di = dinv[i];
    const float s = di * di;
    const float4 hv = ((const float4*)(h + (long long)i * F))[c];
    const float4 bv = ((const float4*)b)[c];
    float4 av = ((const float4*)(agg + (long long)i * F))[c];
    av.x += hv.x * s + bv.x;
    av.y += hv.y * s + bv.y;
    av.z += hv.z * s + bv.z;
    av.w += hv.w * s + bv.w;
    if (RELU) {
        av.x = fmaxf(av.x, 0.f);
        av.y = fmaxf(av.y, 0.f);
        av.z = fmaxf(av.z, 0.f);
        av.w = fmaxf(av.w, 0.f);
    }
    ((float4*)(agg + (long long)i * F))[c] = av;
}

// ---------------- launch ----------------
extern "C" void kernel_launch(void* const* d_in, const int* in_sizes, int n_in,
                              void* d_out, int out_size, void* d_ws, size_t ws_size,
                              hipStream_t stream) {
    const float*     x   = (const float*)d_in[0];
    const long long* ei  = (const long long*)d_in[1];  // int64 [2, E]
    const float*     W1  = (const float*)d_in[2];
    const float*     b1  = (const float*)d_in[3];
    const float*     W2  = (const float*)d_in[4];
    const float*     b2  = (const float*)d_in[5];
    float*           out = (float*)d_out;

    const int IN_F = 128, HID = 64, NC = 16;
    const int N = in_sizes[0] / IN_F;
    const int E = in_sizes[1] / 2;
    const long long* src = ei;
    const long long* dst = ei + E;

    // workspace layout (floats): dinv | h1 | agg1 | h2
    float* ws   = (float*)d_ws;
    size_t off  = 0;
    float* dinv = ws + off; off += ((size_t)N + 127) & ~(size_t)127;
    float* h1   = ws + off; off += (size_t)N * HID;
    float* agg1 = ws + off; off += (size_t)N * HID;
    float* h2   = ws + off; off += (size_t)N * NC;

    const int T = 256;
    // init: deg starts at 1 (self-loop), agg1 and out start at 0
    fill_kernel<<<2048, T, 0, stream>>>(dinv, 1.0f, N);
    fill_kernel<<<4096, T, 0, stream>>>(agg1, 0.0f, N * HID);
    fill_kernel<<<2048, T, 0, stream>>>(out, 0.0f, N * NC);

    degree_kernel<<<(E + T - 1) / T, T, 0, stream>>>(dst, dinv, E);
    rsqrt_kernel<<<(N + T - 1) / T, T, 0, stream>>>(dinv, N);

    const int mtiles = (N + 15) / 16;

    // layer 1
    gemm_wmma_f32<128, 64><<<(mtiles * (64 / 16) + 7) / 8, T, 0, stream>>>(x, W1, h1, N, mtiles);
    edge_scatter<16, 64><<<((long long)E * 16 + T - 1) / T, T, 0, stream>>>(src, dst, dinv, h1, agg1, E);
    self_bias<64, true><<<((long long)N * 16 + T - 1) / T, T, 0, stream>>>(agg1, h1, dinv, b1, N);

    // layer 2 (agg1 now holds ReLU'd features)
    gemm_wmma_f32<64, 16><<<(mtiles + 7) / 8, T, 0, stream>>>(agg1, W2, h2, N, mtiles);
    edge_scatter<4, 16><<<((long long)E * 4 + T - 1) / T, T, 0, stream>>>(src, dst, dinv, h2, out, E);
    self_bias<16, false><<<((long long)N * 4 + T - 1) / T, T, 0, stream>>>(out, h2, dinv, b2, N);
}